// NeuroPredessor_30270929502576
// MI455X (gfx1250) — compile-verified
//
#include <hip/hip_runtime.h>
#include <hip/hip_bf16.h>
#include <math.h>

#define DIM     128
#define NVARS   12000
#define NNODES  8400
#define NROUNDS 12
#define NNZ_CAP 262144
#define BSTRIDE 132   // padded LDS row stride (floats): 132%64=4 -> conflict-free B reads

typedef __attribute__((ext_vector_type(2))) float v2f;
typedef __attribute__((ext_vector_type(8))) float v8f;

__device__ __forceinline__ float sigm_(float x) { return 1.0f / (1.0f + expf(-x)); }

// Stage one 128-row x 128-col fp32 weight chunk (row-major, K contiguous) into LDS
// at byte offset ldsBase, padded to BSTRIDE floats per row, via async-to-LDS DMA.
// 128 threads; each issues 32 coalesced 16B transfers (ASYNCcnt <= 32 per wave).
__device__ __forceinline__ void stage_weights_async(const float* __restrict__ Wchunk,
                                                    unsigned ldsBase, int tid) {
#pragma unroll
  for (int i = 0; i < 32; ++i) {
    const int seg = (i << 7) + tid;  // 0..4095 : row = seg>>5, 16B segment = seg&31
    const unsigned lo =
        ldsBase + ((((unsigned)(seg >> 5)) * (unsigned)BSTRIDE + ((unsigned)(seg & 31) << 2)) << 2);
    const float* gp = Wchunk + ((size_t)seg << 2);
    asm volatile("global_load_async_to_lds_b128 %0, %1, off" ::"v"(lo), "v"(gp) : "memory");
  }
}

// ---------------------------------------------------------------------------
// GEMM: Y[M,N] = act(X[M,128] @ W[N,128]^T + bias)
// block = 128 threads (4 waves), grid = (M/16, N/128).
// Weight chunk staged in LDS (async); each wave computes a 16x32 tile with
// 4 independent V_WMMA_F32_16X16X4_F32 accumulator chains (K split + N split).
// Fragment layout per CDNA5 ISA: A/B: lane = 16*(K>=2) + (M|N), vgpr = K&1.
//   C/D: vgpr v, lanes<16 -> (M=v, N=lane), lanes>=16 -> (M=v+8, N=lane-16).
// ---------------------------------------------------------------------------
template <int ACT>  // 0 = none, 1 = relu
__global__ __launch_bounds__(128) void gemm_xwt(const float* __restrict__ X,
                                                const float* __restrict__ W,
                                                const float* __restrict__ bias,
                                                float* __restrict__ Y, int N) {
  extern __shared__ float smem[];
  const int tid   = threadIdx.x;
  const int lane  = tid & 31;
  const int wave  = tid >> 5;                 // 0..3
  const int m0    = blockIdx.x << 4;
  const int nbase = blockIdx.y << 7;

  stage_weights_async(W + ((size_t)nbase << 7), (unsigned)(uintptr_t)smem, tid);
  asm volatile("s_wait_asynccnt 0x0" ::: "memory");
  __syncthreads();

  const int koff = (lane >> 4) << 1;          // 0 or 2
  const float* xr  = X + (size_t)(m0 + (lane & 15)) * DIM + koff;
  const float* b0p = smem + ((wave << 5) + (lane & 15)) * BSTRIDE + koff;
  const float* b1p = b0p + 16 * BSTRIDE;

  v8f a00 = {}, a01 = {}, a10 = {}, a11 = {};
#pragma unroll
  for (int k = 0; k < DIM; k += 8) {
    const v2f a = *(const v2f*)(xr + k);
    a00 = __builtin_amdgcn_wmma_f32_16x16x4_f32(false, a, false, *(const v2f*)(b0p + k),
                                                (short)0, a00, false, false);
    a10 = __builtin_amdgcn_wmma_f32_16x16x4_f32(false, a, false, *(const v2f*)(b1p + k),
                                                (short)0, a10, false, false);
    const v2f a2 = *(const v2f*)(xr + k + 4);
    a01 = __builtin_amdgcn_wmma_f32_16x16x4_f32(false, a2, false, *(const v2f*)(b0p + k + 4),
                                                (short)0, a01, false, false);
    a11 = __builtin_amdgcn_wmma_f32_16x16x4_f32(false, a2, false, *(const v2f*)(b1p + k + 4),
                                                (short)0, a11, false, false);
  }
  const v8f d0 = a00 + a01;
  const v8f d1 = a10 + a11;

  const int col0  = nbase + (wave << 5) + (lane & 15);
  const int col1  = col0 + 16;
  const int rbase = m0 + ((lane >> 4) << 3);
  const float bv0 = bias[col0], bv1 = bias[col1];
#pragma unroll
  for (int v = 0; v < 8; ++v) {
    float x0 = d0[v] + bv0, x1 = d1[v] + bv1;
    if (ACT) { x0 = fmaxf(x0, 0.0f); x1 = fmaxf(x1, 0.0f); }
    Y[(size_t)(rbase + v) * N + col0] = x0;
    Y[(size_t)(rbase + v) * N + col1] = x1;
  }
}

// Dual-input GEMM for LSTM gates: G[M,512] = X1@W1^T + X2@W2^T + b1 + b2
// Both weight chunks staged in LDS; 4 accumulator chains carried across both phases.
__global__ __launch_bounds__(128) void gemm_lstm(const float* __restrict__ X1,
                                                 const float* __restrict__ W1,
                                                 const float* __restrict__ X2,
                                                 const float* __restrict__ W2,
                                                 const float* __restrict__ b1,
                                                 const float* __restrict__ b2,
                                                 float* __restrict__ G) {
  extern __shared__ float smem[];
  const int tid   = threadIdx.x;
  const int lane  = tid & 31;
  const int wave  = tid >> 5;
  const int m0    = blockIdx.x << 4;
  const int nbase = blockIdx.y << 7;

  const unsigned sbase = (unsigned)(uintptr_t)smem;
  stage_weights_async(W1 + ((size_t)nbase << 7), sbase, tid);
  stage_weights_async(W2 + ((size_t)nbase << 7), sbase + (unsigned)(128 * BSTRIDE * 4), tid);
  asm volatile("s_wait_asynccnt 0x0" ::: "memory");
  __syncthreads();

  const int koff = (lane >> 4) << 1;
  const float* x1r = X1 + (size_t)(m0 + (lane & 15)) * DIM + koff;
  const float* x2r = X2 + (size_t)(m0 + (lane & 15)) * DIM + koff;
  const float* s1  = smem + ((wave << 5) + (lane & 15)) * BSTRIDE + koff;
  const float* s2  = s1 + 128 * BSTRIDE;

  v8f a00 = {}, a01 = {}, a10 = {}, a11 = {};
#pragma unroll
  for (int k = 0; k < DIM; k += 8) {
    const v2f a = *(const v2f*)(x1r + k);
    a00 = __builtin_amdgcn_wmma_f32_16x16x4_f32(false, a, false, *(const v2f*)(s1 + k),
                                                (short)0, a00, false, false);
    a10 = __builtin_amdgcn_wmma_f32_16x16x4_f32(false, a, false, *(const v2f*)(s1 + 16 * BSTRIDE + k),
                                                (short)0, a10, false, false);
    const v2f a2 = *(const v2f*)(x1r + k + 4);
    a01 = __builtin_amdgcn_wmma_f32_16x16x4_f32(false, a2, false, *(const v2f*)(s1 + k + 4),
                                                (short)0, a01, false, false);
    a11 = __builtin_amdgcn_wmma_f32_16x16x4_f32(false, a2, false, *(const v2f*)(s1 + 16 * BSTRIDE + k + 4),
                                                (short)0, a11, false, false);
  }
#pragma unroll
  for (int k = 0; k < DIM; k += 8) {
    const v2f a = *(const v2f*)(x2r + k);
    a00 = __builtin_amdgcn_wmma_f32_16x16x4_f32(false, a, false, *(const v2f*)(s2 + k),
                                                (short)0, a00, false, false);
    a10 = __builtin_amdgcn_wmma_f32_16x16x4_f32(false, a, false, *(const v2f*)(s2 + 16 * BSTRIDE + k),
                                                (short)0, a10, false, false);
    const v2f a2 = *(const v2f*)(x2r + k + 4);
    a01 = __builtin_amdgcn_wmma_f32_16x16x4_f32(false, a2, false, *(const v2f*)(s2 + k + 4),
                                                (short)0, a01, false, false);
    a11 = __builtin_amdgcn_wmma_f32_16x16x4_f32(false, a2, false, *(const v2f*)(s2 + 16 * BSTRIDE + k + 4),
                                                (short)0, a11, false, false);
  }
  const v8f d0 = a00 + a01;
  const v8f d1 = a10 + a11;

  const int col0  = nbase + (wave << 5) + (lane & 15);
  const int col1  = col0 + 16;
  const int rbase = m0 + ((lane >> 4) << 3);
  const float bv0 = b1[col0] + b2[col0];
  const float bv1 = b1[col1] + b2[col1];
#pragma unroll
  for (int v = 0; v < 8; ++v) {
    G[(size_t)(rbase + v) * 512 + col0] = d0[v] + bv0;
    G[(size_t)(rbase + v) * 512 + col1] = d1[v] + bv1;
  }
}

// ---------------------------------------------------------------------------
// Sparse (0/1-valued) matmul: Y[row,:] = sum over nnz of X[idx,:]. Wave/row.
// ---------------------------------------------------------------------------
__global__ __launch_bounds__(256) void spmm(const int* __restrict__ ptr,
                                            const int* __restrict__ idx,
                                            const float* __restrict__ X,
                                            float* __restrict__ Y, int rows) {
  const int row = (blockIdx.x << 3) + (threadIdx.x >> 5);
  if (row >= rows) return;
  const int lane = threadIdx.x & 31;
  const int beg = ptr[row], end = ptr[row + 1];
  float4 acc = make_float4(0.f, 0.f, 0.f, 0.f);
  for (int k = beg; k < end; ++k) {
    const float4 v = *(const float4*)(X + (size_t)idx[k] * DIM + (lane << 2));
    acc.x += v.x; acc.y += v.y; acc.z += v.z; acc.w += v.w;
  }
  *(float4*)(Y + (size_t)row * DIM + (lane << 2)) = acc;
}

// ---------------------------------------------------------------------------
// CSR/CSC construction for the dense 0/1 unpack matrix (deterministic order)
// ---------------------------------------------------------------------------
__global__ void zero_i32(int* __restrict__ p, int n) {
  const int i = blockIdx.x * 256 + threadIdx.x;
  if (i < n) p[i] = 0;
}

__global__ __launch_bounds__(256) void csr_count(const float* __restrict__ U,
                                                 int* __restrict__ rcnt,
                                                 int* __restrict__ ccnt) {
  const int row = (blockIdx.x << 3) + (threadIdx.x >> 5);
  if (row >= NNODES) return;
  const int lane = threadIdx.x & 31;
  const float* u = U + (size_t)row * NVARS;
  int cnt = 0;
  for (int j = 0; j < NVARS; j += 32) {
    const bool nz = u[j + lane] != 0.0f;
    const unsigned m = (unsigned)__ballot(nz);
    cnt += __popc(m);
    if (nz) atomicAdd(&ccnt[j + lane], 1);
  }
  if (lane == 0) rcnt[row] = cnt;
}

__global__ void exscan(const int* __restrict__ cnt, int* __restrict__ optr, int n) {
  __shared__ int part[256];
  const int t = threadIdx.x;
  const int chunk = (n + 255) >> 8;
  const int lo = t * chunk;
  const int hi = min(lo + chunk, n);
  int s = 0;
  for (int i = lo; i < hi; ++i) s += cnt[i];
  part[t] = s;
  __syncthreads();
  if (t == 0) {
    int run = 0;
    for (int i = 0; i < 256; ++i) { int v = part[i]; part[i] = run; run += v; }
    optr[n] = run;
  }
  __syncthreads();
  int run = part[t];
  for (int i = lo; i < hi; ++i) { optr[i] = run; run += cnt[i]; }
}

__global__ __launch_bounds__(256) void csr_fill_rows(const float* __restrict__ U,
                                                     const int* __restrict__ rptr,
                                                     int* __restrict__ cidx) {
  const int row = (blockIdx.x << 3) + (threadIdx.x >> 5);
  if (row >= NNODES) return;
  const int lane = threadIdx.x & 31;
  const float* u = U + (size_t)row * NVARS;
  int base = rptr[row];
  const unsigned lt = (1u << lane) - 1u;
  for (int j = 0; j < NVARS; j += 32) {
    const bool nz = u[j + lane] != 0.0f;
    const unsigned m = (unsigned)__ballot(nz);
    if (nz) cidx[base + __popc(m & lt)] = j + lane;
    base += __popc(m);
  }
}

__global__ __launch_bounds__(256) void csr_fill_cols(const float* __restrict__ U,
                                                     const int* __restrict__ cptr,
                                                     int* __restrict__ ridx) {
  const int col = blockIdx.x * 256 + threadIdx.x;
  if (col >= NVARS) return;
  int pos = cptr[col];
  for (int r = 0; r < NNODES; ++r)
    if (U[(size_t)r * NVARS + col] != 0.0f) ridx[pos++] = r;
}

// ---------------------------------------------------------------------------
// State init and LSTM pointwise update
// ---------------------------------------------------------------------------
__global__ void init_state(const int* __restrict__ vt, const float* __restrict__ tw,
                           const float* __restrict__ tb, const float* __restrict__ fw,
                           const float* __restrict__ fb, float* __restrict__ h,
                           float* __restrict__ c) {
  const int i = blockIdx.x * 256 + threadIdx.x;
  if (i >= NVARS * DIM) return;
  const int r = i >> 7, d = i & 127;
  h[i] = (vt[r] == 1) ? (tw[d] + tb[d]) : (fw[d] + fb[d]);
  c[i] = 0.0f;
}

__global__ void lstm_update(const float* __restrict__ G, float* __restrict__ h,
                            float* __restrict__ c, int M) {
  const int i = blockIdx.x * 256 + threadIdx.x;
  if (i >= M * DIM) return;
  const int r = i >> 7, d = i & 127;
  const float* g = G + (size_t)r * 512;
  const float gi = g[d], gf = g[128 + d], gg = g[256 + d], go = g[384 + d];
  const float c2 = sigm_(gf) * c[i] + sigm_(gi) * tanhf(gg);
  c[i] = c2;
  h[i] = sigm_(go) * tanhf(c2);
}

__global__ __launch_bounds__(256) void vote_out(const float* __restrict__ X,
                                                const float* __restrict__ w3,
                                                const float* __restrict__ b3,
                                                float* __restrict__ out, int M) {
  const int row = (blockIdx.x << 3) + (threadIdx.x >> 5);
  if (row >= M) return;
  const int lane = threadIdx.x & 31;
  const float4 x = *(const float4*)(X + (size_t)row * DIM + (lane << 2));
  const float4 w = *(const float4*)(w3 + (lane << 2));
  float s = x.x * w.x + x.y * w.y + x.z * w.z + x.w * w.w;
  for (int off = 16; off > 0; off >>= 1) s += __shfl_down(s, off);
  if (lane == 0) out[row] = s + b3[0];
}

// ---------------------------------------------------------------------------
extern "C" void kernel_launch(void* const* d_in, const int* in_sizes, int n_in,
                              void* d_out, int out_size, void* d_ws, size_t ws_size,
                              hipStream_t stream) {
  (void)in_sizes; (void)n_in; (void)out_size; (void)ws_size;
  const int*   vt      = (const int*)d_in[0];
  const float* unpack  = (const float*)d_in[1];
  const float* true_w  = (const float*)d_in[2];
  const float* true_b  = (const float*)d_in[3];
  const float* false_w = (const float*)d_in[4];
  const float* false_b = (const float*)d_in[5];
  const float* cm[6]; const float* pm[6]; const float* vv[6];
  for (int i = 0; i < 6; ++i) cm[i] = (const float*)d_in[6 + i];
  for (int i = 0; i < 6; ++i) pm[i] = (const float*)d_in[12 + i];
  for (int i = 0; i < 6; ++i) vv[i] = (const float*)d_in[18 + i];
  const float* vu_wih = (const float*)d_in[24];
  const float* vu_whh = (const float*)d_in[25];
  const float* vu_bih = (const float*)d_in[26];
  const float* vu_bhh = (const float*)d_in[27];
  const float* nu_wih = (const float*)d_in[28];
  const float* nu_whh = (const float*)d_in[29];
  const float* nu_bih = (const float*)d_in[30];
  const float* nu_bhh = (const float*)d_in[31];

  // workspace layout (floats then ints), ~58 MB total
  float* h  = (float*)d_ws;
  float* c  = h  + (size_t)NVARS * DIM;
  float* t0 = c  + (size_t)NVARS * DIM;
  float* t1 = t0 + (size_t)NVARS * DIM;
  float* mb = t1 + (size_t)NVARS * DIM;      // msg / p2c buffer
  float* G  = mb + (size_t)NVARS * DIM;      // LSTM gates [NVARS,512]
  int* rptr = (int*)(G + (size_t)NVARS * 512);
  int* cptr = rptr + (NNODES + 1);
  int* rcnt = cptr + (NVARS + 1);
  int* ccnt = rcnt + NNODES;
  int* cidx = ccnt + NVARS;
  int* ridx = cidx + NNZ_CAP;

  const dim3 b256(256);
  const dim3 b128(128);
  const size_t smem1 = (size_t)128 * BSTRIDE * 4;      // one staged weight chunk
  const size_t smem2 = 2 * smem1;                      // two chunks (LSTM)

  // ---- build CSR (child->parent) and CSC (parent->child) once per launch ----
  zero_i32<<<dim3((NVARS + 255) / 256), b256, 0, stream>>>(ccnt, NVARS);
  csr_count<<<dim3((NNODES + 7) / 8), b256, 0, stream>>>(unpack, rcnt, ccnt);
  exscan<<<dim3(1), b256, 0, stream>>>(rcnt, rptr, NNODES);
  exscan<<<dim3(1), b256, 0, stream>>>(ccnt, cptr, NVARS);
  csr_fill_rows<<<dim3((NNODES + 7) / 8), b256, 0, stream>>>(unpack, rptr, cidx);
  csr_fill_cols<<<dim3((NVARS + 255) / 256), b256, 0, stream>>>(unpack, cptr, ridx);

  init_state<<<dim3((NVARS * DIM + 255) / 256), b256, 0, stream>>>(
      vt, true_w, true_b, false_w, false_b, h, c);

  for (int round = 0; round < NROUNDS; ++round) {
    // children -> parents message: cm MLP on full h, then SpMM
    gemm_xwt<1><<<dim3(NVARS / 16, 1), b128, smem1, stream>>>(h,  cm[0], cm[1], t0, DIM);
    gemm_xwt<1><<<dim3(NVARS / 16, 1), b128, smem1, stream>>>(t0, cm[2], cm[3], t1, DIM);
    gemm_xwt<0><<<dim3(NVARS / 16, 1), b128, smem1, stream>>>(t1, cm[4], cm[5], t0, DIM);
    spmm<<<dim3((NNODES + 7) / 8), b256, 0, stream>>>(rptr, cidx, t0, mb, NNODES);
    // vu LSTM on nodes
    gemm_lstm<<<dim3(NNODES / 16, 4), b128, smem2, stream>>>(mb, vu_wih, h, vu_whh,
                                                             vu_bih, vu_bhh, G);
    lstm_update<<<dim3((NNODES * DIM + 255) / 256), b256, 0, stream>>>(G, h, c, NNODES);
    // parents -> children message: pm MLP on h[:NNODES], then transposed SpMM
    gemm_xwt<1><<<dim3(NNODES / 16, 1), b128, smem1, stream>>>(h,  pm[0], pm[1], t0, DIM);
    gemm_xwt<1><<<dim3(NNODES / 16, 1), b128, smem1, stream>>>(t0, pm[2], pm[3], t1, DIM);
    gemm_xwt<0><<<dim3(NNODES / 16, 1), b128, smem1, stream>>>(t1, pm[4], pm[5], t0, DIM);
    spmm<<<dim3((NVARS + 7) / 8), b256, 0, stream>>>(cptr, ridx, t0, mb, NVARS);
    // nu LSTM on all rows
    gemm_lstm<<<dim3(NVARS / 16, 4), b128, smem2, stream>>>(mb, nu_wih, h, nu_whh,
                                                            nu_bih, nu_bhh, G);
    lstm_update<<<dim3((NVARS * DIM + 255) / 256), b256, 0, stream>>>(G, h, c, NVARS);
  }

  // vote MLP on the variable rows h[NNODES:]
  const int MV = NVARS - NNODES;  // 3600
  gemm_xwt<1><<<dim3(MV / 16, 1), b128, smem1, stream>>>(h + (size_t)NNODES * DIM,
                                                         vv[0], vv[1], t0, DIM);
  gemm_xwt<1><<<dim3(MV / 16, 1), b128, smem1, stream>>>(t0, vv[2], vv[3], t1, DIM);
  vote_out<<<dim3((MV + 7) / 8), b256, 0, stream>>>(t1, vv[4], vv[5], (float*)d_out, MV);
}